// Decoder_6932077216375
// MI455X (gfx1250) — compile-verified
//
#include <hip/hip_runtime.h>
#include <hip/hip_bf16.h>
#include <cstdint>

// Problem dims
#define BB   32
#define TDEC 64
#define UU   32
#define TK   64
#define DD   512
#define VV   50257

// d_out layout (floats), concatenated in reference return order
#define O_VOCAB 0
#define O_TATTN 1608224          // B*V
#define O_PGEN  1610272          // + B*TK
#define O_UCOV  1610304          // + B
#define O_NTC   1611328          // + B*U
#define O_UIDX  1742400          // + 2*U*B*TK

typedef __attribute__((ext_vector_type(2))) float v2f;
typedef __attribute__((ext_vector_type(8))) float v8f;

__device__ __forceinline__ float wave_sum(float v) {
    for (int off = 16; off > 0; off >>= 1) v += __shfl_down(v, off, 32);
    return v;
}
__device__ __forceinline__ float wave_max(float v) {
    for (int off = 16; off > 0; off >>= 1) v = fmaxf(v, __shfl_down(v, off, 32));
    return v;
}
// 256-thread block reductions (sbuf must hold >= 8 floats)
__device__ __forceinline__ float block_sum(float v, float* sbuf) {
    int lane = threadIdx.x & 31, w = threadIdx.x >> 5;
    v = wave_sum(v);
    if (lane == 0) sbuf[w] = v;
    __syncthreads();
    float r = 0.f;
    if (threadIdx.x < 8) r = sbuf[threadIdx.x];
    if (threadIdx.x < 32) r = wave_sum(r);
    if (threadIdx.x == 0) sbuf[0] = r;
    __syncthreads();
    float o = sbuf[0];
    __syncthreads();
    return o;
}
__device__ __forceinline__ float block_max(float v, float* sbuf) {
    int lane = threadIdx.x & 31, w = threadIdx.x >> 5;
    v = wave_max(v);
    if (lane == 0) sbuf[w] = v;
    __syncthreads();
    float r = -3.4e38f;
    if (threadIdx.x < 8) r = sbuf[threadIdx.x];
    if (threadIdx.x < 32) r = wave_max(r);
    if (threadIdx.x == 0) sbuf[0] = r;
    __syncthreads();
    float o = sbuf[0];
    __syncthreads();
    return o;
}

// ---- K1: LayerNorm of last timestep -> local_dec (B,D) ----
__global__ void k_ln(const float* __restrict__ tf, const float* __restrict__ g,
                     const float* __restrict__ bvec, float* __restrict__ ldec) {
    __shared__ float sbuf[8];
    int b = blockIdx.x;
    const float* x = tf + ((size_t)b * TDEC + (TDEC - 1)) * DD;
    float s = 0.f;
    for (int d = threadIdx.x; d < DD; d += 256) s += x[d];
    float mu = block_sum(s, sbuf) * (1.f / DD);
    float vv = 0.f;
    for (int d = threadIdx.x; d < DD; d += 256) { float dx = x[d] - mu; vv += dx * dx; }
    float var = block_sum(vv, sbuf) * (1.f / DD);
    float rstd = rsqrtf(var + 1e-5f);
    for (int d = threadIdx.x; d < DD; d += 256)
        ldec[b * DD + d] = (x[d] - mu) * rstd * g[d] + bvec[d];
}

// ---- K2: dec_u / dec_t = local_dec @ Wp.T + bp ----
__global__ void k_proj(const float* __restrict__ ldec,
                       const float* __restrict__ Wu, const float* __restrict__ bu,
                       const float* __restrict__ Wt, const float* __restrict__ bt,
                       float* __restrict__ du, float* __restrict__ dt) {
    int tid = blockIdx.x * 256 + threadIdx.x;     // 0..16383
    const float* W   = blockIdx.y ? Wt : Wu;
    const float* bia = blockIdx.y ? bt : bu;
    float*       out = blockIdx.y ? dt : du;
    int j = tid >> 5, b = tid & 31;
    const float* wr = W + (size_t)j * DD;
    const float* xr = ldec + b * DD;
    float acc = 0.f;
    for (int d = 0; d < DD; ++d) acc += xr[d] * wr[d];
    out[b * DD + j] = acc + bia[j];
}

// ---- K3: utterance attention: u_out, u_attn, u_cov, utter_index ----
__global__ void k_uatt(const float* __restrict__ du, const float* __restrict__ enc,
                       const float* __restrict__ umask, const float* __restrict__ cov,
                       const float* __restrict__ v, const float* __restrict__ wc,
                       float* __restrict__ u_out, float* __restrict__ u_attn_ws,
                       int* __restrict__ uidx_ws, float* __restrict__ out) {
    __shared__ float sscore[32];
    __shared__ float sattn[32];
    int b = blockIdx.x;
    int lane = threadIdx.x & 31, wv = threadIdx.x >> 5;
    const float* dec = du + b * DD;
    for (int uu = 0; uu < 4; ++uu) {
        int u = wv * 4 + uu;
        const float* e = enc + ((size_t)b * UU + u) * DD;
        float c = cov[b * UU + u];
        float p = 0.f;
        for (int d = lane; d < DD; d += 32)
            p += tanhf(e[d] + dec[d] + c * wc[d]) * v[d];
        p = wave_sum(p);
        if (lane == 0) sscore[u] = p;
    }
    __syncthreads();
    if (threadIdx.x < 32) {
        float s = sscore[lane];
        float m = wave_max(s); m = __shfl(m, 0, 32);
        float e = expf(s - m);
        float tot = wave_sum(e); tot = __shfl(tot, 0, 32);
        float a_ = (e / tot) * umask[b * UU + lane];
        float nf = wave_sum(a_); nf = __shfl(nf, 0, 32);
        float attn = a_ / (nf == 0.f ? 1.f : nf);
        sattn[lane] = attn;
        u_attn_ws[b * UU + lane] = attn;
        out[O_UCOV + b * UU + lane] = cov[b * UU + lane] + attn;
    }
    __syncthreads();
    if (threadIdx.x == 0) {
        int best = 0; float bv = sattn[0];
        for (int u = 1; u < UU; ++u) if (sattn[u] > bv) { bv = sattn[u]; best = u; }
        uidx_ws[b] = best;
        out[O_UIDX + b] = (float)best;
    }
    __syncthreads();
    for (int d = threadIdx.x; d < DD; d += 256) {
        float acc = 0.f;
        for (int u = 0; u < UU; ++u) acc += sattn[u] * enc[((size_t)b * UU + u) * DD + d];
        u_out[b * DD + d] = acc;
    }
}

// ---- K4: p_gen ----
__global__ void k_pgen(const float* __restrict__ u_out, const float* __restrict__ ldec,
                       const float* __restrict__ W, const float* __restrict__ bias,
                       float* __restrict__ out) {
    __shared__ float sbuf[8];
    int b = blockIdx.x;
    float p = 0.f;
    for (int k = threadIdx.x; k < 2 * DD; k += 256) {
        float x = (k < DD) ? u_out[b * DD + k] : ldec[b * DD + (k - DD)];
        p += x * W[k];
    }
    float tot = block_sum(p, sbuf);
    if (threadIdx.x == 0)
        out[O_PGEN + b] = 1.f / (1.f + expf(-(tot + bias[0])));
}

// ---- K5: token attention scores + softmax -> attn, next_tok_cov ----
// Stages dec_t row / ta_v / ta_wc into LDS via async global->LDS (CDNA5).
__global__ void k_tatt(const float* __restrict__ dt, const float* __restrict__ tokf,
                       const float* __restrict__ tmask, const float* __restrict__ tcov,
                       const float* __restrict__ v, const float* __restrict__ wc,
                       float* __restrict__ attn_ws, float* __restrict__ out) {
    __shared__ float sdec[DD];
    __shared__ float sv[DD];
    __shared__ float swc[DD];
    __shared__ float sscore[TK];
    int ub = blockIdx.x;              // u*B + b
    int b = ub & 31;
    int lane = threadIdx.x & 31, wv = threadIdx.x >> 5;

    // async stage 1536 floats (3 x 512) as 384 x b128 per-lane copies
    for (int i = threadIdx.x; i < 384; i += 256) {
        const float* src; float* dst;
        if (i < 128)      { src = dt + b * DD + i * 4;     dst = &sdec[i * 4]; }
        else if (i < 256) { src = v + (i - 128) * 4;       dst = &sv[(i - 128) * 4]; }
        else              { src = wc + (i - 256) * 4;      dst = &swc[(i - 256) * 4]; }
        unsigned ldsaddr = (unsigned)(uintptr_t)dst;       // low 32b of flat = LDS offset
        unsigned long long ga = (unsigned long long)(uintptr_t)src;
        asm volatile("global_load_async_to_lds_b128 %0, %1, off"
                     :: "v"(ldsaddr), "v"(ga) : "memory");
    }
    asm volatile("s_wait_asynccnt 0" ::: "memory");
    __syncthreads();

    for (int t = wv; t < TK; t += 8) {
        float c = tcov[(size_t)ub * TK + t];
        const float* tf = tokf + ((size_t)ub * TK + t) * DD;
        float acc = 0.f;
        for (int d = lane * 4; d < DD; d += 128) {
            float4 tv = *(const float4*)(tf + d);
            acc += tanhf(tv.x + sdec[d + 0] + c * swc[d + 0]) * sv[d + 0];
            acc += tanhf(tv.y + sdec[d + 1] + c * swc[d + 1]) * sv[d + 1];
            acc += tanhf(tv.z + sdec[d + 2] + c * swc[d + 2]) * sv[d + 2];
            acc += tanhf(tv.w + sdec[d + 3] + c * swc[d + 3]) * sv[d + 3];
        }
        acc = wave_sum(acc);
        if (lane == 0) sscore[t] = acc;
    }
    __syncthreads();
    if (threadIdx.x < 32) {
        float s0 = sscore[lane], s1 = sscore[lane + 32];
        float m = wave_max(fmaxf(s0, s1)); m = __shfl(m, 0, 32);
        float e0 = expf(s0 - m), e1 = expf(s1 - m);
        float tot = wave_sum(e0 + e1); tot = __shfl(tot, 0, 32);
        float a0 = (e0 / tot) * tmask[(size_t)ub * TK + lane];
        float a1 = (e1 / tot) * tmask[(size_t)ub * TK + lane + 32];
        float nf = wave_sum(a0 + a1); nf = __shfl(nf, 0, 32);
        nf = (nf == 0.f) ? 1.f : nf;
        a0 /= nf; a1 /= nf;
        attn_ws[(size_t)ub * TK + lane]      = a0;
        attn_ws[(size_t)ub * TK + lane + 32] = a1;
        out[O_NTC + (size_t)ub * TK + lane]      = tcov[(size_t)ub * TK + lane] + a0;
        out[O_NTC + (size_t)ub * TK + lane + 32] = tcov[(size_t)ub * TK + lane + 32] + a1;
    }
}

// ---- K6: selected token weighted sum -> target_out, target_attn_dist ----
__global__ void k_sel(const int* __restrict__ uidx, const float* __restrict__ attn_ws,
                      const float* __restrict__ tokf, float* __restrict__ tout,
                      float* __restrict__ out) {
    __shared__ float sattn[TK];
    int b = blockIdx.x;
    int u = uidx[b];
    size_t ub = (size_t)u * BB + b;
    if (threadIdx.x < TK) {
        float a = attn_ws[ub * TK + threadIdx.x];
        sattn[threadIdx.x] = a;
        out[O_TATTN + b * TK + threadIdx.x] = a;
    }
    __syncthreads();
    for (int d = threadIdx.x; d < DD; d += 256) {
        float acc = 0.f;
        const float* base = tokf + ub * TK * DD + d;
        for (int t = 0; t < TK; ++t) acc += sattn[t] * base[t * DD];
        tout[b * DD + d] = acc;
    }
}

// ---- K7: h = concat(target_out, local_dec) @ out1_W.T + out1_b ----
__global__ void k_out1(const float* __restrict__ tout, const float* __restrict__ ldec,
                       const float* __restrict__ W, const float* __restrict__ bias,
                       float* __restrict__ h) {
    int tid = blockIdx.x * 256 + threadIdx.x;
    int j = tid >> 5, b = tid & 31;
    const float* wr = W + (size_t)j * (2 * DD);
    const float* x0 = tout + b * DD;
    const float* x1 = ldec + b * DD;
    float acc = 0.f;
    for (int k = 0; k < DD; ++k) acc += x0[k] * wr[k];
    for (int k = 0; k < DD; ++k) acc += x1[k] * wr[DD + k];
    h[b * DD + j] = acc + bias[j];
}

// ---- K8: logits = h @ out2_W.T + out2_b via V_WMMA_F32_16X16X4_F32 ----
// Wave = one 16-wide N tile, both M tiles (rows 0-15, 16-31) share the B frag.
// h staged in LDS with row stride 516 (pad 4 floats) for conflict-free ds reads.
#define HPAD 516
__global__ void k_vocab(const float* __restrict__ h, const float* __restrict__ W,
                        const float* __restrict__ bias, float* __restrict__ logits) {
    extern __shared__ float sh[];     // 32 * 516 floats
    for (int i = threadIdx.x; i < (BB * DD) / 4; i += 256) {
        int gi = i * 4;
        int row = gi >> 9, col = gi & (DD - 1);
        *(float4*)(sh + row * HPAD + col) = *(const float4*)(h + gi);
    }
    __syncthreads();
    int wave = threadIdx.x >> 5, lane = threadIdx.x & 31;
    const int NT = (VV + 15) / 16;    // 3142
    int tile = blockIdx.x * 8 + wave;
    if (tile >= NT) return;           // wave-uniform
    int n = tile * 16 + (lane & 15);
    int kh = (lane >> 4) * 2;         // K sub-offset per lane half
    int m = lane & 15;
    bool valid = n < VV;
    const float* wrow = W + (size_t)(valid ? n : 0) * DD;
    v8f acc0 = {}; v8f acc1 = {};
    for (int k = 0; k < DD; k += 4) {
        float2 wv2 = *(const float2*)(wrow + k + kh);
        v2f bf; bf[0] = wv2.x; bf[1] = wv2.y;
        float2 h0 = *(const float2*)(sh + m * HPAD + k + kh);
        float2 h1 = *(const float2*)(sh + (16 + m) * HPAD + k + kh);
        v2f a0; a0[0] = h0.x; a0[1] = h0.y;
        v2f a1; a1[0] = h1.x; a1[1] = h1.y;
        acc0 = __builtin_amdgcn_wmma_f32_16x16x4_f32(false, a0, false, bf,
                                                     (short)0, acc0, false, false);
        acc1 = __builtin_amdgcn_wmma_f32_16x16x4_f32(false, a1, false, bf,
                                                     (short)0, acc1, false, false);
    }
    if (!valid) return;
    float bv = bias[n];
    int mbase = (lane < 16) ? 0 : 8;
    for (int r = 0; r < 8; ++r) {
        int row = mbase + r;
        logits[(size_t)row * VV + n]        = acc0[r] + bv;
        logits[(size_t)(16 + row) * VV + n] = acc1[r] + bv;
    }
}

// ---- K9: per-row max & sum(exp) over V ----
__global__ void k_red(const float* __restrict__ logits, float* __restrict__ rmax,
                      float* __restrict__ rsum) {
    __shared__ float sbuf[8];
    int b = blockIdx.x;
    const float* row = logits + (size_t)b * VV;
    float m = -3.4e38f;
    for (int i = threadIdx.x; i < VV; i += 256) m = fmaxf(m, row[i]);
    m = block_max(m, sbuf);
    float s = 0.f;
    for (int i = threadIdx.x; i < VV; i += 256) s += expf(row[i] - m);
    s = block_sum(s, sbuf);
    if (threadIdx.x == 0) { rmax[b] = m; rsum[b] = s; }
}

// ---- K10: in-place softmax normalize ----
__global__ void k_norm(float* __restrict__ logits, const float* __restrict__ rmax,
                       const float* __restrict__ rsum) {
    size_t i = (size_t)blockIdx.x * 256 + threadIdx.x;
    if (i >= (size_t)BB * VV) return;
    int b = (int)(i / VV);
    logits[i] = expf(logits[i] - rmax[b]) / rsum[b];
}

extern "C" void kernel_launch(void* const* d_in, const int* in_sizes, int n_in,
                              void* d_out, int out_size, void* d_ws, size_t ws_size,
                              hipStream_t stream) {
    const float* tf    = (const float*)d_in[0];   // target_features
    const float* enc   = (const float*)d_in[2];   // encode_features
    const float* umask = (const float*)d_in[3];
    const float* tokf  = (const float*)d_in[4];   // token_features
    const float* tmask = (const float*)d_in[5];
    const float* cov   = (const float*)d_in[6];
    const float* tcov  = (const float*)d_in[7];
    const float* uaWp  = (const float*)d_in[8];
    const float* uabp  = (const float*)d_in[9];
    const float* uav   = (const float*)d_in[10];
    const float* uawc  = (const float*)d_in[11];
    const float* taWp  = (const float*)d_in[12];
    const float* tabp  = (const float*)d_in[13];
    const float* tav   = (const float*)d_in[14];
    const float* tawc  = (const float*)d_in[15];
    const float* pgW   = (const float*)d_in[16];
    const float* pgb   = (const float*)d_in[17];
    const float* o1W   = (const float*)d_in[18];
    const float* o1b   = (const float*)d_in[19];
    const float* o2W   = (const float*)d_in[20];
    const float* o2b   = (const float*)d_in[21];
    const float* lng   = (const float*)d_in[22];
    const float* lnb   = (const float*)d_in[23];

    float* out = (float*)d_out;
    float* ws = (float*)d_ws;
    float* w_ldec  = ws;                 // 16384
    float* w_du    = ws + 16384;         // 16384
    float* w_dt    = ws + 32768;         // 16384
    float* w_uout  = ws + 49152;         // 16384
    float* w_uattn = ws + 65536;         // 1024
    int*   w_uidx  = (int*)(ws + 66560); // 32
    float* w_tout  = ws + 66592;         // 16384
    float* w_h     = ws + 82976;         // 16384
    float* w_attn  = ws + 99360;         // 131072
    float* w_rmax  = ws + 230432;        // 32
    float* w_rsum  = ws + 230464;        // 32

    k_ln<<<BB, 256, 0, stream>>>(tf, lng, lnb, w_ldec);
    k_proj<<<dim3(64, 2), 256, 0, stream>>>(w_ldec, uaWp, uabp, taWp, tabp, w_du, w_dt);
    k_uatt<<<BB, 256, 0, stream>>>(w_du, enc, umask, cov, uav, uawc,
                                   w_uout, w_uattn, w_uidx, out);
    k_pgen<<<BB, 256, 0, stream>>>(w_uout, w_ldec, pgW, pgb, out);
    k_tatt<<<UU * BB, 256, 0, stream>>>(w_dt, tokf, tmask, tcov, tav, tawc, w_attn, out);
    k_sel<<<BB, 256, 0, stream>>>(w_uidx, w_attn, tokf, w_tout, out);
    k_out1<<<64, 256, 0, stream>>>(w_tout, w_ldec, o1W, o1b, w_h);
    k_vocab<<<393, 256, BB * HPAD * sizeof(float), stream>>>(w_h, o2W, o2b, out + O_VOCAB);
    k_red<<<BB, 256, 0, stream>>>(out + O_VOCAB, w_rmax, w_rsum);
    k_norm<<<(BB * VV + 255) / 256, 256, 0, stream>>>(out + O_VOCAB, w_rmax, w_rsum);
}